// Generator_44203803411098
// MI455X (gfx1250) — compile-verified
//
#include <hip/hip_runtime.h>
#include <hip/hip_bf16.h>

// Problem constants from the reference.
#define ENTITIES_N  100000
#define RELATIONS_N 1000
#define OUT_WIDTH   (2 * ENTITIES_N + RELATIONS_N)   // 201000 floats per row
#define OUT_WIDTH4  (OUT_WIDTH / 4)                  // 50250 float4s per row (201000 % 4 == 0)
#define BATCH       1024
#define BLOCK       256
#define UNROLL      4                                 // float4s per thread

typedef float v4f __attribute__((ext_vector_type(4)));

// Fused zero-fill + one-hot scatter, 4x unrolled.
//
// Roofline: output is 1024 x 201000 fp32 = 823 MB of pure stores -> ~35 us at
// 23.3 TB/s HBM. No FLOPs of substance, so WMMA is inapplicable; the CDNA5
// features that matter are wave32 + non-temporal B128 stores (output is 4.3x
// the 192 MB L2, so we must stream past it).
//
// Each thread stores UNROLL float4s strided by BLOCK, so every store
// instruction is a fully coalesced 512 B/wave burst. The three hot columns are
// uniform per block (scalar loads); the fast path does only a float4-granular
// compare (3 v_cmp) and the 12-compare element fixup runs on at most 3 lanes
// per *row* (out of 50250 float4s), i.e. essentially never.
__global__ __launch_bounds__(BLOCK) void onehot_rows_kernel(
    const int* __restrict__ hID,
    const int* __restrict__ rID,
    const int* __restrict__ tID,
    float* __restrict__ out)
{
    const int row = (int)blockIdx.y;

    // Uniform per block: hot element columns and their float4 indices.
    const int c0 = hID[row];                              // [0, 100000)
    const int c1 = ENTITIES_N + rID[row];                 // [100000, 101000)
    const int c2 = ENTITIES_N + RELATIONS_N + tID[row];   // [101000, 201000)
    const int h4 = c0 >> 2;
    const int r4 = c1 >> 2;
    const int t4 = c2 >> 2;

    float* rowp = out + (size_t)row * OUT_WIDTH;

    const int base4 = (int)blockIdx.x * (BLOCK * UNROLL) + (int)threadIdx.x;

#pragma unroll
    for (int k = 0; k < UNROLL; ++k) {
        const int col4 = base4 + k * BLOCK;
        if (col4 < OUT_WIDTH4) {
            v4f v = {0.0f, 0.0f, 0.0f, 0.0f};
            // Rare path: this float4 contains one of the three hot elements.
            if (col4 == h4 || col4 == r4 || col4 == t4) {
                const int col = col4 * 4;
                v.x = ((col + 0) == c0 || (col + 0) == c1 || (col + 0) == c2) ? 1.0f : 0.0f;
                v.y = ((col + 1) == c0 || (col + 1) == c1 || (col + 1) == c2) ? 1.0f : 0.0f;
                v.z = ((col + 2) == c0 || (col + 2) == c1 || (col + 2) == c2) ? 1.0f : 0.0f;
                v.w = ((col + 3) == c0 || (col + 3) == c1 || (col + 3) == c2) ? 1.0f : 0.0f;
            }
            // 16B-aligned streaming store -> global_store_b128 th:TH_NT.
            __builtin_nontemporal_store(v, (v4f*)(rowp + (size_t)col4 * 4));
        }
    }
}

extern "C" void kernel_launch(void* const* d_in, const int* in_sizes, int n_in,
                              void* d_out, int out_size, void* d_ws, size_t ws_size,
                              hipStream_t stream) {
    // Inputs (setup_inputs order): z [1024*128] f32 (unused), hID [1024] i32,
    // rID [1024] i32, tID [1024] i32. Output: 1024*201000 f32.
    (void)in_sizes; (void)n_in; (void)out_size; (void)d_ws; (void)ws_size;

    const int* hID = (const int*)d_in[1];
    const int* rID = (const int*)d_in[2];
    const int* tID = (const int*)d_in[3];
    float* out = (float*)d_out;

    const int blocks_x = (OUT_WIDTH4 + BLOCK * UNROLL - 1) / (BLOCK * UNROLL);  // 50
    dim3 grid(blocks_x, BATCH, 1);   // (50, 1024)
    dim3 block(BLOCK, 1, 1);
    onehot_rows_kernel<<<grid, block, 0, stream>>>(hID, rID, tID, out);
}